// DiscriminativeLoss_12979391169049
// MI455X (gfx1250) — compile-verified
//
#include <hip/hip_runtime.h>
#include <math.h>

typedef __attribute__((ext_vector_type(2))) float v2f;
typedef __attribute__((ext_vector_type(8))) float v8f;

#define NCLS 65          // labels 0..64 (0 = background)
#define NCT 5            // class tiles of 16 -> covers 80 >= 65 (pad classes never match)
#define P1_BLOCKS 256
#define P2_BLOCKS 1024
#define THREADS 256

// ---------------------------------------------------------------------------
// Pass 1: per-class embedding sums via one-hot x embeddings WMMA (f32 16x16x4),
// per-class counts via deterministic integer LDS histogram.
// A (16x4): one-hot of labels for 16 classes x 4 points.
//   VGPR0: lanes 0-15 -> (M=lane, K=0)=point p0 ; lanes 16-31 -> K=2 = p2
//   VGPR1: K=1 = p1 ; K=3 = p3
// B (4x16): embeddings, VGPR0: K=0 (lanes 0-15) / K=2 (lanes 16-31), VGPR1: K=1/K=3
// C/D (16x16 f32): VGPR v, lanes 0-15 -> M=v ; lanes 16-31 -> M=v+8 ; N=lane&15
// Group loop is forced wave-uniform (readfirstlane) so EXEC is all-ones at
// every WMMA (ISA 7.12 requirement) and loop control + label loads go scalar.
// ---------------------------------------------------------------------------
__global__ __launch_bounds__(THREADS)
void dl_pass1(const float* __restrict__ emb, const int* __restrict__ labels, int M,
              float* __restrict__ psums, int* __restrict__ pcounts)
{
    __shared__ float smem[NCT * 16 * 32];   // 80 classes x 32 dims = 10 KB
    __shared__ int   hist[NCLS];

    const int tid  = threadIdx.x;
    const int lane = tid & 31;
    const int wave = tid >> 5;
    const int half = lane >> 4;
    const int l16  = lane & 15;

    for (int i = tid; i < NCT * 16 * 32; i += THREADS) smem[i] = 0.0f;
    for (int i = tid; i < NCLS; i += THREADS) hist[i] = 0;
    __syncthreads();

    v8f acc[NCT][2];
    #pragma unroll
    for (int t = 0; t < NCT; ++t)
        #pragma unroll
        for (int n = 0; n < 2; ++n)
            acc[t][n] = (v8f){0.f,0.f,0.f,0.f,0.f,0.f,0.f,0.f};

    const int M4 = M >> 2;
    // provably wave-uniform loop bounds -> scalar loop control, EXEC untouched
    const int waveGlobal = __builtin_amdgcn_readfirstlane((blockIdx.x * THREADS + tid) >> 5);
    const int totalWaves = (gridDim.x * THREADS) >> 5;

    for (int g = waveGlobal; g < M4; g += totalWaves) {
        const int p = g << 2;                 // scalar
        const int lab0 = labels[p + 0];       // uniform address -> s_load
        const int lab1 = labels[p + 1];
        const int lab2 = labels[p + 2];
        const int lab3 = labels[p + 3];
        if (lane < 4) atomicAdd(&hist[labels[p + lane]], 1);   // integer: deterministic

        // B tiles (embeddings of the 4 points)
        const float* r02 = emb + (size_t)(p + 2 * half) * 32;  // K=0 / K=2 row
        const float* r13 = r02 + 32;                            // K=1 / K=3 row
        v2f b0, b1;
        b0.x = r02[l16];        b0.y = r13[l16];        // dims 0..15
        b1.x = r02[l16 + 16];   b1.y = r13[l16 + 16];   // dims 16..31

        const int la02 = half ? lab2 : lab0;
        const int la13 = half ? lab3 : lab1;

        #pragma unroll
        for (int t = 0; t < NCT; ++t) {
            v2f a;
            a.x = (la02 == t * 16 + l16) ? 1.0f : 0.0f;
            a.y = (la13 == t * 16 + l16) ? 1.0f : 0.0f;
            acc[t][0] = __builtin_amdgcn_wmma_f32_16x16x4_f32(
                false, a, false, b0, (short)0, acc[t][0], false, false);
            acc[t][1] = __builtin_amdgcn_wmma_f32_16x16x4_f32(
                false, a, false, b1, (short)0, acc[t][1], false, false);
        }
    }

    // Wave-serial (fixed-order, deterministic) reduction of accumulators into LDS
    for (int w = 0; w < THREADS / 32; ++w) {
        if (wave == w) {
            #pragma unroll
            for (int t = 0; t < NCT; ++t)
                #pragma unroll
                for (int n = 0; n < 2; ++n)
                    #pragma unroll
                    for (int v = 0; v < 8; ++v) {
                        const int cls = t * 16 + v + 8 * half;
                        const int dim = n * 16 + l16;
                        smem[cls * 32 + dim] += acc[t][n][v];
                    }
        }
        __syncthreads();
    }

    // Tail (M not multiple of 4): handled serially by block 0, race-free
    const int tail = M - (M4 << 2);
    if (blockIdx.x == 0 && tail > 0) {
        if (tid < 32) {
            for (int tp = 0; tp < tail; ++tp) {
                const int pt  = (M4 << 2) + tp;
                const int lab = labels[pt];
                smem[lab * 32 + tid] += emb[(size_t)pt * 32 + tid];
            }
        }
        if (tid == 0)
            for (int tp = 0; tp < tail; ++tp) hist[labels[(M4 << 2) + tp]] += 1;
        __syncthreads();
    }

    for (int i = tid; i < NCLS * 32; i += THREADS)
        psums[(size_t)blockIdx.x * (NCLS * 32) + i] = smem[i];
    for (int i = tid; i < NCLS; i += THREADS)
        pcounts[blockIdx.x * NCLS + i] = hist[i];
}

// ---------------------------------------------------------------------------
// Combine per-block partials (fixed order) -> means + float counts
// ---------------------------------------------------------------------------
__global__ void dl_reduce_means(const float* __restrict__ psums, const int* __restrict__ pcounts,
                                float* __restrict__ means, float* __restrict__ counts_f)
{
    const int idx = blockIdx.x * blockDim.x + threadIdx.x;
    if (idx >= NCLS * 32) return;
    const int c = idx >> 5;
    float s = 0.0f;
    for (int b = 0; b < P1_BLOCKS; ++b) s += psums[(size_t)b * (NCLS * 32) + idx];
    int cnt = 0;
    for (int b = 0; b < P1_BLOCKS; ++b) cnt += pcounts[b * NCLS + c];
    const float sc = (cnt > 0) ? (float)cnt : 1.0f;
    means[idx] = s / sc;
    if ((idx & 31) == 0) counts_f[c] = (float)cnt;
}

// ---------------------------------------------------------------------------
// Pass 2: per-point pull hinge; pull = (1/C) * sum_i pull_i / count[label_i]
// Each block writes one deterministic partial.
// ---------------------------------------------------------------------------
__global__ __launch_bounds__(THREADS)
void dl_pass2(const float* __restrict__ emb, const int* __restrict__ labels, int M,
              const float* __restrict__ means, const float* __restrict__ counts_f,
              float* __restrict__ ppull)
{
    __shared__ float sm[NCLS * 32];
    __shared__ float sc[NCLS];
    __shared__ float red[THREADS];
    const int tid = threadIdx.x;
    for (int i = tid; i < NCLS * 32; i += THREADS) sm[i] = means[i];
    for (int i = tid; i < NCLS; i += THREADS) {
        const float c = counts_f[i];
        sc[i] = (c > 0.0f) ? c : 1.0f;
    }
    __syncthreads();

    float local = 0.0f;
    const int stride = gridDim.x * THREADS;
    for (int i = blockIdx.x * THREADS + tid; i < M; i += stride) {
        const int lab = labels[i];
        if (lab > 0) {
            const float4* e4 = (const float4*)(emb + (size_t)i * 32);
            const float* mv = &sm[lab * 32];
            float d2 = 0.0f;
            #pragma unroll
            for (int k = 0; k < 8; ++k) {
                const float4 v = e4[k];
                const float dx = v.x - mv[k * 4 + 0];
                const float dy = v.y - mv[k * 4 + 1];
                const float dz = v.z - mv[k * 4 + 2];
                const float dw = v.w - mv[k * 4 + 3];
                d2 += dx * dx + dy * dy + dz * dz + dw * dw;
            }
            const float dist = sqrtf(d2);
            const float h = fmaxf(dist - 0.5f, 0.0f);
            local += (h * h) / sc[lab];
        }
    }
    red[tid] = local;
    __syncthreads();
    for (int s = THREADS >> 1; s > 0; s >>= 1) {
        if (tid < s) red[tid] += red[tid + s];
        __syncthreads();
    }
    if (tid == 0) ppull[blockIdx.x] = red[0];
}

// ---------------------------------------------------------------------------
// Finalize: C, pull, O(65^2) push, reg, total. Single block, deterministic.
// ---------------------------------------------------------------------------
__global__ __launch_bounds__(THREADS)
void dl_finalize(const float* __restrict__ means, const float* __restrict__ counts_f,
                 const float* __restrict__ ppull, int np, float* __restrict__ out)
{
    __shared__ float sm[NCLS * 32];
    __shared__ int   pres[NCLS];
    __shared__ float redf[THREADS];
    __shared__ int   redi[THREADS];
    __shared__ float pullSum;
    const int tid = threadIdx.x;
    for (int i = tid; i < NCLS * 32; i += THREADS) sm[i] = means[i];
    if (tid < NCLS) pres[tid] = (tid > 0 && counts_f[tid] > 0.0f) ? 1 : 0;
    __syncthreads();

    // C = number of present instances
    redi[tid] = (tid < NCLS) ? pres[tid] : 0;
    __syncthreads();
    for (int s = THREADS >> 1; s > 0; s >>= 1) {
        if (tid < s) redi[tid] += redi[tid + s];
        __syncthreads();
    }
    const int C = redi[0];
    __syncthreads();

    if (tid == 0) {                     // fixed-order pull partial sum
        float s = 0.0f;
        for (int i = 0; i < np; ++i) s += ppull[i];
        pullSum = s;
    }

    // push over upper-triangular pairs
    float pushAcc = 0.0f;
    int pairAcc = 0;
    const int totalPairs = NCLS * (NCLS - 1) / 2;
    for (int pi = tid; pi < totalPairs; pi += THREADS) {
        int i = 0, rem = pi;
        while (rem >= NCLS - 1 - i) { rem -= NCLS - 1 - i; ++i; }
        const int j = i + 1 + rem;
        if (pres[i] && pres[j]) {
            float d2 = 0.0f;
            #pragma unroll 8
            for (int e = 0; e < 32; ++e) {
                const float d = sm[i * 32 + e] - sm[j * 32 + e];
                d2 += d * d;
            }
            const float pd = sqrtf(d2);
            const float t = fmaxf(3.0f - pd, 0.0f);   // 2*DELTA_PUSH = 3.0
            pushAcc += t * t;
            pairAcc += 1;
        }
    }
    redf[tid] = pushAcc; redi[tid] = pairAcc;
    __syncthreads();
    for (int s = THREADS >> 1; s > 0; s >>= 1) {
        if (tid < s) { redf[tid] += redf[tid + s]; redi[tid] += redi[tid + s]; }
        __syncthreads();
    }
    const float pushSum = redf[0];
    const int nPairs = redi[0];
    __syncthreads();

    // regularizer
    float r = 0.0f;
    if (tid < NCLS && pres[tid]) {
        float m2 = 0.0f;
        #pragma unroll 8
        for (int e = 0; e < 32; ++e) { const float v = sm[tid * 32 + e]; m2 += v * v; }
        r = sqrtf(m2);
    }
    redf[tid] = r;
    __syncthreads();
    for (int s = THREADS >> 1; s > 0; s >>= 1) {
        if (tid < s) redf[tid] += redf[tid + s];
        __syncthreads();
    }

    if (tid == 0) {
        const float Cf = (C > 0) ? (float)C : 1.0f;
        const float pull = pullSum / Cf;
        const float push = (nPairs > 0) ? pushSum / (float)nPairs : 0.0f;
        const float reg  = redf[0] / Cf;
        const float total = 1.0f * pull + 1.0f * push + 0.001f * reg;
        out[0] = (C > 0) ? total : 0.0f;
    }
}

// ---------------------------------------------------------------------------
extern "C" void kernel_launch(void* const* d_in, const int* in_sizes, int n_in,
                              void* d_out, int out_size, void* d_ws, size_t ws_size,
                              hipStream_t stream)
{
    (void)n_in; (void)out_size; (void)ws_size;
    const float* emb    = (const float*)d_in[0];
    const int*   labels = (const int*)d_in[1];
    const int    M      = in_sizes[1];

    float* ws       = (float*)d_ws;
    float* psums    = ws;                                              // P1_BLOCKS*65*32
    int*   pcounts  = (int*)(psums + (size_t)P1_BLOCKS * NCLS * 32);   // P1_BLOCKS*65
    float* means    = (float*)(pcounts + P1_BLOCKS * NCLS);            // 65*32
    float* counts_f = means + NCLS * 32;                               // 65
    float* ppull    = counts_f + NCLS;                                 // P2_BLOCKS

    dl_pass1<<<P1_BLOCKS, THREADS, 0, stream>>>(emb, labels, M, psums, pcounts);
    dl_reduce_means<<<(NCLS * 32 + THREADS - 1) / THREADS, THREADS, 0, stream>>>(
        psums, pcounts, means, counts_f);
    dl_pass2<<<P2_BLOCKS, THREADS, 0, stream>>>(emb, labels, M, means, counts_f, ppull);
    dl_finalize<<<1, THREADS, 0, stream>>>(means, counts_f, ppull, P2_BLOCKS, (float*)d_out);
}